// RetrievalModule_69544110457211
// MI455X (gfx1250) — compile-verified
//
#include <hip/hip_runtime.h>
#include <hip/hip_bf16.h>
#include <math.h>

// Problem constants (from reference)
#define T_TAB   8
#define N_ROWS  500000
#define D_DIM   64
#define B_BATCH 16384
#define L_BAG   20
#define F_DIM   256

typedef __attribute__((ext_vector_type(2))) float v2f;
typedef __attribute__((ext_vector_type(8))) float v8f;

// ---------------------------------------------------------------------------
// Kernel 1: dense_embs[B,16] = dense_features[B,256] @ W_dense[256,10] + b
// via V_WMMA_F32_16X16X4_F32 (full f32 precision). One wave per 16x16 output
// tile. A-tile staged in LDS (stride 260 -> conflict-free fragment reads).
// B (W_dense, zero-padded 10->16 cols) staged in LDS with K-pairs interleaved
// so each B fragment is one aligned ds_load_b64. All masking is done with
// clamped indices + selects: no EXEC-masked loads in the K-loop.
// ---------------------------------------------------------------------------
#define LDS_STRIDE 260

__global__ __launch_bounds__(32)
void dense_wmma_kernel(const float* __restrict__ dense,
                       const float* __restrict__ Wd,   // [256,10]
                       const float* __restrict__ bd,   // [10]
                       float* __restrict__ de)         // [B,16] workspace
{
    __shared__ __align__(16) float atile[16 * LDS_STRIDE];   // 16.6 KB
    __shared__ __align__(16) float btile[F_DIM * 16];        // 16 KB, padded
    const int lane = threadIdx.x;        // 0..31, full wave (EXEC all 1s)
    const int b0   = blockIdx.x * 16;

    // --- Stage B: W_dense zero-padded to [256][16], pair-interleaved layout.
    // btile[(k>>1)*32 + n*2 + (k&1)] = (n<10) ? Wd[k*10+n] : 0
    for (int i = lane; i < F_DIM * 16; i += 32) {
        const int k  = i >> 4;
        const int n  = i & 15;
        const int nn = (n < 10) ? n : 0;              // clamp: load stays in-bounds
        const float v = Wd[k * 10 + nn];              // unconditional load
        btile[(k >> 1) * 32 + n * 2 + (k & 1)] = (n < 10) ? v : 0.0f;  // select
    }

    // --- Stage A: 16x256 tile, 512 B coalesced per row.
    for (int r = 0; r < 16; ++r) {
        const float4* src = (const float4*)(dense + (size_t)(b0 + r) * F_DIM);
        float4 x = src[lane];
        float4 y = src[lane + 32];
        *(float4*)&atile[r * LDS_STRIDE + lane * 4]       = x;
        *(float4*)&atile[r * LDS_STRIDE + 128 + lane * 4] = y;
    }
    __syncthreads();

    const int m  = lane & 15;            // A row / B-D col index
    const int kk = (lane >> 4) * 2;      // lanes 16-31 cover K=+2,+3
    const int n  = m;

    v8f c = {};
#pragma unroll 4
    for (int k0 = 0; k0 < F_DIM; k0 += 4) {
        const float2 av = *(const float2*)&atile[m * LDS_STRIDE + k0 + kk];
        const float2 bv = *(const float2*)&btile[((k0 + kk) >> 1) * 32 + n * 2];
        v2f a;  a.x = av.x;  a.y = av.y;
        v2f b;  b.x = bv.x;  b.y = bv.y;
        c = __builtin_amdgcn_wmma_f32_16x16x4_f32(
                false, a, false, b, (short)0, c, false, false);
    }

    const int   nb    = (n < 10) ? n : 0;
    const float bias  = (n < 10) ? bd[nb] : 0.0f;
    const int   mbase = (lane >> 4) * 8; // C/D: VGPR r -> M = r (+8 hi half)
#pragma unroll
    for (int r = 0; r < 8; ++r)
        de[(size_t)(b0 + mbase + r) * 16 + n] = c[r] + bias;
}

// ---------------------------------------------------------------------------
// Kernel 2: fused gather + pooled dot + dense dot + sigmoid.
// logit[b] = sum_{t,l} dot(table[t, idx[b,t,l], :], W_over[t*64:(t+1)*64])
//          + dot(dense_embs[b,0:10], W_over[512:522]) + b_over
// One wave per bag row, 2 rows per wave -> 8192 waves. Each 256 B embedding
// row is two coalesced 128 B lines (float2 per lane). L=20 fully unrolled so
// 20 index loads then 20 row gathers per table are simultaneously in flight.
// ---------------------------------------------------------------------------
#define ROWS_PER_WAVE 2
#define WAVES_PER_BLK 8

__global__ __launch_bounds__(256)
void fused_main_kernel(const int*   __restrict__ idx,     // [B,T,L]
                       const float* __restrict__ de,      // [B,16]
                       const float* __restrict__ tab,     // [T,N,D]
                       const float* __restrict__ Wover,   // [522]
                       const float* __restrict__ bover,   // [1]
                       float* __restrict__ out)           // d_out (preds at +1)
{
    const int lane = threadIdx.x & 31;
    const int wave = threadIdx.x >> 5;
    const int wgid = blockIdx.x * WAVES_PER_BLK + wave;

    // W_over sparse slices: 2 floats/lane/table, resident in 16 VGPRs.
    float2 w[T_TAB];
#pragma unroll
    for (int t = 0; t < T_TAB; ++t)
        w[t] = *(const float2*)(Wover + t * D_DIM + 2 * lane);
    const float wdense = (lane < 10) ? Wover[T_TAB * D_DIM + lane] : 0.0f;
    const float bo     = bover[0];

    for (int r = 0; r < ROWS_PER_WAVE; ++r) {
        const int b = wgid * ROWS_PER_WAVE + r;
        float acc = 0.0f;
#pragma unroll
        for (int t = 0; t < T_TAB; ++t) {
            const int*   ip = idx + ((size_t)b * T_TAB + t) * L_BAG;
            const float* tb = tab + (size_t)t * N_ROWS * D_DIM;
            int n[L_BAG];
#pragma unroll
            for (int l = 0; l < L_BAG; ++l)           // 20 idx loads in flight
                n[l] = ip[l];
            float2 rv[L_BAG];
#pragma unroll
            for (int l = 0; l < L_BAG; ++l)           // 20 gathers in flight
                rv[l] = *(const float2*)(tb + (size_t)n[l] * D_DIM + 2 * lane);
#pragma unroll
            for (int l = 0; l < L_BAG; ++l) {
                acc = fmaf(rv[l].x, w[t].x, acc);
                acc = fmaf(rv[l].y, w[t].y, acc);
            }
        }
        // dense contribution: lanes 0..9 carry it, others weighted by 0
        const float deval = de[(size_t)b * 16 + (lane & 15)];
        acc = fmaf(deval, wdense, acc);

        // wave32 reduction
#pragma unroll
        for (int off = 16; off > 0; off >>= 1)
            acc += __shfl_xor(acc, off, 32);

        if (lane == 0) {
            const float logit = acc + bo;
            out[1 + b] = 1.0f / (1.0f + __expf(-logit));
        }
    }
}

// ---------------------------------------------------------------------------
// Kernel 3: deterministic sum of preds -> d_out[0] (single block, fixed tree).
// ---------------------------------------------------------------------------
__global__ __launch_bounds__(256)
void reduce_sum_kernel(const float* __restrict__ preds, float* __restrict__ out0)
{
    __shared__ float s[256];
    float a = 0.0f;
    for (int i = threadIdx.x; i < B_BATCH; i += 256)
        a += preds[i];
    s[threadIdx.x] = a;
    __syncthreads();
    for (int off = 128; off > 0; off >>= 1) {
        if (threadIdx.x < off) s[threadIdx.x] += s[threadIdx.x + off];
        __syncthreads();
    }
    if (threadIdx.x == 0) out0[0] = s[0];
}

// ---------------------------------------------------------------------------
extern "C" void kernel_launch(void* const* d_in, const int* in_sizes, int n_in,
                              void* d_out, int out_size, void* d_ws, size_t ws_size,
                              hipStream_t stream)
{
    const int*   indices = (const int*)  d_in[0];   // [B,T,L]
    const float* densef  = (const float*)d_in[1];   // [B,F]
    const float* tables  = (const float*)d_in[2];   // [T,N,D]
    const float* Wd      = (const float*)d_in[3];   // [F,10]
    const float* bd      = (const float*)d_in[4];   // [10]
    const float* Wo      = (const float*)d_in[5];   // [522,1]
    const float* bo      = (const float*)d_in[6];   // [1]

    float* out = (float*)d_out;                     // [0]=sum, [1..B]=pred
    float* de  = (float*)d_ws;                      // [B,16] dense_embs

    // 1) dense_embs via f32 WMMA: 1024 tiles of 16 rows, 1 wave each.
    dense_wmma_kernel<<<B_BATCH / 16, 32, 0, stream>>>(densef, Wd, bd, de);

    // 2) fused gather / pool / overarch / sigmoid: 8192 waves.
    fused_main_kernel<<<B_BATCH / (WAVES_PER_BLK * ROWS_PER_WAVE), 256, 0, stream>>>(
        indices, de, tables, Wo, bo, out);

    // 3) deterministic scalar reduction.
    reduce_sum_kernel<<<1, 256, 0, stream>>>(out + 1, out);
}